// HSEModule_17506286698927
// MI455X (gfx1250) — compile-verified
//
#include <hip/hip_runtime.h>
#include <hip/hip_bf16.h>

// CDNA5 / gfx1250 Squeeze-and-Excitation:
//   pooled = mean(x, HW); h = relu(pooled @ w1^T + b1);
//   s = hardsigmoid(h @ w2^T + b2); out = x * s
// Memory-bound: x = 154 MB, fits the 192 MB L2. Pool pass warms L2 with x
// (default RT loads); scale pass reads x as last-use NT and streams out with
// NT stores so the 154 MB of output writes do not evict x from L2.
// Effective HBM traffic ~308 MB -> ~13 us @ 23.3 TB/s.
// GEMMs are tiny (38 MFLOP) -> full fp32 precision V_WMMA_F32_16X16X4_F32.

typedef __attribute__((ext_vector_type(2))) float v2f;
typedef __attribute__((ext_vector_type(4))) float v4f;
typedef __attribute__((ext_vector_type(8))) float v8f;

#define BATCH   64
#define C_DIM   768
#define HID     192
#define HW      784           // 28*28
#define HW4     196           // HW / 4 (float4s per row)
#define ROWS    (BATCH * C_DIM)

// ---------------------------------------------------------------------------
// Kernel 1: global average pool. One wave per (b,c) row of 784 floats.
// Coalesced b128 loads (default temporal -> fills L2 for the scale pass),
// wave32 shuffle reduction.
// ---------------------------------------------------------------------------
__global__ void se_pool(const float* __restrict__ x, float* __restrict__ pooled) {
  const int row  = (int)((blockIdx.x * blockDim.x + threadIdx.x) >> 5);
  const int lane = (int)(threadIdx.x & 31);
  const v4f* p = reinterpret_cast<const v4f*>(x + (size_t)row * HW);
  float s = 0.0f;
  for (int i = lane; i < HW4; i += 32) {
    v4f v = p[i];
    s += (v.x + v.y) + (v.z + v.w);
  }
  #pragma unroll
  for (int off = 16; off > 0; off >>= 1) s += __shfl_xor(s, off, 32);
  if (lane == 0) pooled[row] = s * (1.0f / (float)HW);
}

// ---------------------------------------------------------------------------
// Kernel 2: h = relu(pooled @ w1^T + b1)   [64,192] = [64,768] x [192,768]^T
// One wave per 16x16 D tile; K looped in steps of 4 (V_WMMA_F32_16X16X4_F32).
// A(16x4): lanes 0-15 hold K={0,1}, lanes 16-31 hold K={2,3} (VGPR pair).
// B(4x16): same K split across lane halves; N = lane & 15.
// D(16x16): VGPR r -> row r + 8*(lane>=16), col lane & 15.
// ---------------------------------------------------------------------------
__global__ void se_gemm1_wmma(const float* __restrict__ pooled,
                              const float* __restrict__ w1,
                              const float* __restrict__ b1,
                              float* __restrict__ h) {
  const int wave = (int)((blockIdx.x * blockDim.x + threadIdx.x) >> 5); // 0..47
  const int lane = (int)(threadIdx.x & 31);
  const int mt = wave / (HID / 16);       // 0..3
  const int nt = wave % (HID / 16);       // 0..11
  const int m0 = mt * 16, n0 = nt * 16;
  const int half = lane >> 4;             // 0: K={0,1}, 1: K={2,3}
  const int l15  = lane & 15;

  const float* arow = pooled + (size_t)(m0 + l15) * C_DIM + 2 * half;
  const float* brow = w1     + (size_t)(n0 + l15) * C_DIM + 2 * half;

  v8f acc = {};
  #pragma unroll 4
  for (int k0 = 0; k0 < C_DIM; k0 += 4) {
    v2f a = *reinterpret_cast<const v2f*>(arow + k0);   // b64 load
    v2f b = *reinterpret_cast<const v2f*>(brow + k0);   // b64 load
    acc = __builtin_amdgcn_wmma_f32_16x16x4_f32(
        /*neg_a=*/false, a, /*neg_b=*/false, b,
        /*c_mod=*/(short)0, acc, /*reuse_a=*/false, /*reuse_b=*/false);
  }

  const float bias = b1[n0 + l15];
  #pragma unroll
  for (int r = 0; r < 8; ++r) {
    float v = acc[r] + bias;
    v = v > 0.0f ? v : 0.0f;                            // ReLU
    h[(size_t)(m0 + r + 8 * half) * HID + (n0 + l15)] = v;
  }
}

// ---------------------------------------------------------------------------
// Kernel 3: s = hardsigmoid(h @ w2^T + b2)  [64,768] = [64,192] x [768,192]^T
// ---------------------------------------------------------------------------
__global__ void se_gemm2_wmma(const float* __restrict__ h,
                              const float* __restrict__ w2,
                              const float* __restrict__ b2,
                              float* __restrict__ s) {
  const int wave = (int)((blockIdx.x * blockDim.x + threadIdx.x) >> 5); // 0..191
  const int lane = (int)(threadIdx.x & 31);
  const int mt = wave / (C_DIM / 16);     // 0..3
  const int nt = wave % (C_DIM / 16);     // 0..47
  const int m0 = mt * 16, n0 = nt * 16;
  const int half = lane >> 4;
  const int l15  = lane & 15;

  const float* arow = h  + (size_t)(m0 + l15) * HID + 2 * half;
  const float* brow = w2 + (size_t)(n0 + l15) * HID + 2 * half;

  v8f acc = {};
  #pragma unroll 4
  for (int k0 = 0; k0 < HID; k0 += 4) {
    v2f a = *reinterpret_cast<const v2f*>(arow + k0);
    v2f b = *reinterpret_cast<const v2f*>(brow + k0);
    acc = __builtin_amdgcn_wmma_f32_16x16x4_f32(
        false, a, false, b, (short)0, acc, false, false);
  }

  const float bias = b2[n0 + l15];
  #pragma unroll
  for (int r = 0; r < 8; ++r) {
    float v = (acc[r] + bias) * (1.0f / 6.0f) + 0.5f;   // hardsigmoid
    v = v < 0.0f ? 0.0f : (v > 1.0f ? 1.0f : v);
    s[(size_t)(m0 + r + 8 * half) * C_DIM + (n0 + l15)] = v;
  }
}

// ---------------------------------------------------------------------------
// Kernel 4: out = x * s (broadcast over HW). One block per (b,c) row;
// s[row] is a uniform (scalar) load. x read is the LAST use -> non-temporal
// load; out is never re-read -> non-temporal store, so the 154 MB of output
// writes do not evict the L2-resident x.
// ---------------------------------------------------------------------------
__global__ void se_scale(const float* __restrict__ x,
                         const float* __restrict__ s,
                         float* __restrict__ out) {
  const int row = (int)blockIdx.x;          // 0..49151 (uniform)
  const float sc = s[row];                  // s_load (scalar)
  const int i = (int)threadIdx.x;           // 224 threads, 196 active
  if (i < HW4) {
    const v4f* px = reinterpret_cast<const v4f*>(x   + (size_t)row * HW);
    v4f*       po = reinterpret_cast<v4f*>      (out + (size_t)row * HW);
    v4f v = __builtin_nontemporal_load(px + i);         // last use of x
    v *= sc;
    __builtin_nontemporal_store(v, po + i);             // streaming output
  }
}

// ---------------------------------------------------------------------------
// Launch: inputs are (x, w1, b1, w2, b2) fp32. d_ws holds pooled | h | s.
// ---------------------------------------------------------------------------
extern "C" void kernel_launch(void* const* d_in, const int* in_sizes, int n_in,
                              void* d_out, int out_size, void* d_ws, size_t ws_size,
                              hipStream_t stream) {
  const float* x  = (const float*)d_in[0];
  const float* w1 = (const float*)d_in[1];
  const float* b1 = (const float*)d_in[2];
  const float* w2 = (const float*)d_in[3];
  const float* b2 = (const float*)d_in[4];
  float* out = (float*)d_out;

  char* ws = (char*)d_ws;
  float* pooled = (float*)(ws);                               // 64*768*4  = 192 KiB
  float* h      = (float*)(ws + (size_t)ROWS * 4);            // 64*192*4  =  48 KiB
  float* s      = (float*)(ws + (size_t)ROWS * 4 + (size_t)BATCH * HID * 4);

  // 1) pool: 49152 rows, 1 wave each, 8 waves per block
  se_pool<<<ROWS / 8, 256, 0, stream>>>(x, pooled);

  // 2) GEMM1: 4x12 = 48 tiles, 4 waves per block
  se_gemm1_wmma<<<12, 128, 0, stream>>>(pooled, w1, b1, h);

  // 3) GEMM2: 4x48 = 192 tiles, 8 waves per block
  se_gemm2_wmma<<<24, 256, 0, stream>>>(h, w2, b2, s);

  // 4) scale: one block (224 threads) per row
  se_scale<<<ROWS, 224, 0, stream>>>(x, s, out);
}